// GlobalGraph_68899865362922
// MI455X (gfx1250) — compile-verified
//
#include <hip/hip_runtime.h>

// ---------------------------------------------------------------------------
// Masked attention with custom sum-normalization, fp32 end-to-end via
// V_WMMA_F32_16X16X4_F32 on gfx1250 (wave32).
//   H=64 hidden, N=512 objects, B=128 batch.
// Kernel 1 (x3): qp/kp/vp = X @ W^T + b   ([N,B,H] -> [B,N,H])
// Kernel 2: one workgroup per batch; kp[b] & vp[b] staged fully in LDS
//           (320KB WGP LDS) via the Tensor Data Mover (TDM applies the
//           bank-conflict padding in hardware), scores + exp*mask + rowsum +
//           A~ write + A~@vp, then per-row 1/rowsum rescale (L2-hot, b128).
// ---------------------------------------------------------------------------

typedef float v2f __attribute__((ext_vector_type(2)));
typedef float v8f __attribute__((ext_vector_type(8)));
typedef unsigned int v4u __attribute__((ext_vector_type(4)));
typedef int v4i __attribute__((ext_vector_type(4)));
typedef int v8i __attribute__((ext_vector_type(8)));

constexpr int kH = 64;    // hidden
constexpr int kN = 512;   // objects
constexpr int kB = 128;   // batch
constexpr int kKV_STRIDE = 68;  // padded LDS row stride (floats): 64 + 4-DWORD TDM pad
constexpr int kAT_STRIDE = 18;  // per-wave transpose tile stride, even

// ROCm 7.2 (clang-22) has the 5-arg tensor_load_to_lds builtin; clang-23's
// 6-arg variant is uncharacterized, so fall back to a cooperative copy there.
#if defined(__has_builtin)
#if __has_builtin(__builtin_amdgcn_tensor_load_to_lds) && (__clang_major__ < 23)
#define USE_TDM 1
#endif
#endif

static __device__ __forceinline__ v8f wmma_f32_16x16x4(v2f a, v2f b, v8f c) {
    // D(16x16,f32) = A(16x4,f32) * B(4x16,f32) + C
    return __builtin_amdgcn_wmma_f32_16x16x4_f32(false, a, false, b, (short)0, c,
                                                 false, false);
}

#ifdef USE_TDM
// One TDM descriptor moves a 512x64 f32 tile (128KB) global->LDS, inserting
// 4 DWORDs of padding after every 64-DWORD row (ISA 8.3/8.4 bit layout).
static __device__ __forceinline__ void tdm_stage_512x64(const float* gsrc,
                                                        float* lds_dst) {
    const unsigned long long ga = (unsigned long long)(uintptr_t)gsrc;
    const unsigned int lds_off = (unsigned int)(uintptr_t)lds_dst;  // addr[31:0]

    v4u g0;
    g0[0] = 1u;                                      // count=1 valid descriptor
    g0[1] = lds_off;                                 // lds_addr
    g0[2] = (unsigned int)(ga & 0xffffffffu);        // global_addr[31:0]
    g0[3] = (unsigned int)(ga >> 32) | (2u << 30);   // global_addr[56:32] | type=2

    v8i g1;
    g1[0] = (int)((2u << 16)      // data_size = 4 bytes
                | (1u << 20)      // pad_enable
                | (5u << 22)      // pad_interval: 64 DWORDs
                | (3u << 25));    // pad_amount:   4 DWORDs
    g1[1] = (int)((unsigned)kH << 16);   // tensor_dim0 = 64 (bits 79:48)
    g1[2] = (int)((unsigned)kN << 16);   // tensor_dim1 = 512 (bits 111:80)
    g1[3] = (int)((unsigned)kH << 16);   // tile_dim0 = 64 (bits 127:112)
    g1[4] = (int)kN;                     // tile_dim1 = 512, tile_dim2 = 0
    g1[5] = (int)kH;                     // tensor_dim0_stride = 64
    g1[6] = 0;                           // tensor_dim1_stride unused (2D tile)
    g1[7] = 0;

    v4i zz = {0, 0, 0, 0};
    __builtin_amdgcn_tensor_load_to_lds(g0, g1, zz, zz, 0);
}
#endif

// ---------------------------------------------------------------------------
// Projection: Y[g, :] = X[row(g), :] @ W^T + bias, g = b*N + n flat row index,
// input row = n*B + b (handles the [N,B,H] -> [B,N,H] transpose).
// Grid: 512 blocks x 256 thr (8 waves); each wave = one 16-row x 64-col slab.
// ---------------------------------------------------------------------------
__global__ __launch_bounds__(256) void proj_kernel(
    const float* __restrict__ X, const float* __restrict__ W,
    const float* __restrict__ bias, float* __restrict__ Y)
{
    __shared__ float Wl[kH * kKV_STRIDE];  // 64x68
    __shared__ float bl[kH];

    const int tid = threadIdx.x;
    for (int i = tid; i < kH * kH; i += 256)
        Wl[(i >> 6) * kKV_STRIDE + (i & 63)] = W[i];
    if (tid < kH) bl[tid] = bias[tid];
    __syncthreads();

    const int wv   = tid >> 5;
    const int lane = tid & 31;
    const int half = lane >> 4;
    const int r    = lane & 15;

    const int tile = blockIdx.x * 8 + wv;   // 0..4095
    const int R0   = tile * 16;             // flat (b,n) row base
    const int g    = R0 + r;                // this lane's A-frag row
    const float* xrow = X + ((size_t)(g & (kN - 1)) * kB + (g >> 9)) * kH;

    // A-frags: row r of the 16x64 slab, K = 4kk + 2*half + {0,1}
    v2f qa[16];
#pragma unroll
    for (int kk = 0; kk < 16; ++kk)
        qa[kk] = *(const v2f*)(xrow + kk * 4 + half * 2);

#pragma unroll
    for (int ct = 0; ct < 4; ++ct) {
        const int c0 = ct * 16;
        v8f acc = {0.f, 0.f, 0.f, 0.f, 0.f, 0.f, 0.f, 0.f};
#pragma unroll
        for (int kk = 0; kk < 16; ++kk) {
            // B[k_dim=h, n_col=out] = W[out, h]: row (c0+r) of Wl, consecutive h
            v2f bf = *(const v2f*)(&Wl[(c0 + r) * kKV_STRIDE + kk * 4 + half * 2]);
            acc = wmma_f32_16x16x4(qa[kk], bf, acc);
        }
        const float bb = bl[c0 + r];
#pragma unroll
        for (int j = 0; j < 8; ++j)
            Y[(size_t)(R0 + half * 8 + j) * kH + c0 + r] = acc[j] + bb;
    }
}

// ---------------------------------------------------------------------------
// Attention: one workgroup (256 thr = 8 waves) per batch b.
// ---------------------------------------------------------------------------
__global__ __launch_bounds__(256) void attn_kernel(
    const float* __restrict__ qp, const float* __restrict__ kp,
    const float* __restrict__ vp, const float* __restrict__ mask,
    float* __restrict__ out, float* __restrict__ Aout)
{
    __shared__ float kpL[kN * kKV_STRIDE];          // 139,264 B
    __shared__ float vpL[kN * kKV_STRIDE];          // 139,264 B
    __shared__ float atL[8 * 16 * kAT_STRIDE];      //   9,216 B  (287,744 B total)

    const int b   = blockIdx.x;
    const int tid = threadIdx.x;

    // Stage kp[b], vp[b] (each 512x64 f32, 128KB) into padded LDS.
    const float* kpg = kp + (size_t)b * kN * kH;
    const float* vpg = vp + (size_t)b * kN * kH;
#ifdef USE_TDM
    if (tid < 32) {  // wave 0 issues the DMAs; TDM ignores EXEC
        tdm_stage_512x64(kpg, kpL);
        tdm_stage_512x64(vpg, vpL);
        __builtin_amdgcn_s_wait_tensorcnt(0);
    }
#else
    for (int i = tid; i < kN * (kH / 4); i += 256) {
        const int row = i >> 4;
        const int c4  = (i & 15) * 4;
        *(float4*)(&kpL[row * kKV_STRIDE + c4]) = *(const float4*)(kpg + row * kH + c4);
        *(float4*)(&vpL[row * kKV_STRIDE + c4]) = *(const float4*)(vpg + row * kH + c4);
    }
#endif
    __syncthreads();

    const int wv   = tid >> 5;
    const int lane = tid & 31;
    const int half = lane >> 4;
    const int r    = lane & 15;
    float* atw = &atL[wv * 16 * kAT_STRIDE];
    const float* qpb = qp + (size_t)b * kN * kH;

    for (int t = 0; t < 4; ++t) {
        const int n0 = (wv * 4 + t) * 16;

        // Q A-frags for this 16-row tile (held across the whole m loop).
        const float* qrow = qpb + (size_t)(n0 + r) * kH;
        v2f qa[16];
#pragma unroll
        for (int kk = 0; kk < 16; ++kk)
            qa[kk] = *(const v2f*)(qrow + kk * 4 + half * 2);

        v8f oacc[4];
#pragma unroll
        for (int c = 0; c < 4; ++c)
            oacc[c] = (v8f){0.f, 0.f, 0.f, 0.f, 0.f, 0.f, 0.f, 0.f};
        float rs[8] = {0.f, 0.f, 0.f, 0.f, 0.f, 0.f, 0.f, 0.f};

        for (int mt = 0; mt < 32; ++mt) {
            const int m0 = mt * 16;

            // S(16x16) = Q(16x64) @ K^T(64x16): B[k=h, n=m] = kp[m0+m, h]
            v8f s = {0.f, 0.f, 0.f, 0.f, 0.f, 0.f, 0.f, 0.f};
#pragma unroll
            for (int kk = 0; kk < 16; ++kk) {
                v2f bf = *(const v2f*)(&kpL[(m0 + r) * kKV_STRIDE + kk * 4 + half * 2]);
                s = wmma_f32_16x16x4(qa[kk], bf, s);
            }

            // exp(S/8) * mask; stash unnormalized A~; accumulate row sums.
            // C-layout: lane (half,r) holds rows n0+8*half+j, column m0+r.
            const size_t abase = ((size_t)b * kN + n0 + half * 8) * (size_t)kN + m0 + r;
            float e[8];
#pragma unroll
            for (int j = 0; j < 8; ++j) {
                float ev = __expf(s[j] * 0.125f) * mask[abase + (size_t)j * kN];
                e[j] = ev;
                rs[j] += ev;
                Aout[abase + (size_t)j * kN] = ev;               // unnormalized
                atw[(half * 8 + j) * kAT_STRIDE + r] = ev;       // C->A transpose
            }
            __builtin_amdgcn_wave_barrier();

            // out(16x64) += A~(16x16) @ vp_tile(16x64)
#pragma unroll
            for (int kk2 = 0; kk2 < 4; ++kk2) {
                v2f aa = *(const v2f*)(&atw[r * kAT_STRIDE + kk2 * 4 + half * 2]);
                const int kr = m0 + kk2 * 4 + half * 2;
#pragma unroll
                for (int c = 0; c < 4; ++c) {
                    v2f bf;
                    bf.x = vpL[(size_t)kr * kKV_STRIDE + c * 16 + r];
                    bf.y = vpL[(size_t)(kr + 1) * kKV_STRIDE + c * 16 + r];
                    oacc[c] = wmma_f32_16x16x4(aa, bf, oacc[c]);
                }
            }
            __builtin_amdgcn_wave_barrier();
        }

        // Row sums: reduce across the 16 lanes of each half (wave32).
        float invv[8];
#pragma unroll
        for (int j = 0; j < 8; ++j) {
            float v = rs[j];
            v += __shfl_xor(v, 1, 32);
            v += __shfl_xor(v, 2, 32);
            v += __shfl_xor(v, 4, 32);
            v += __shfl_xor(v, 8, 32);
            invv[j] = (v == 0.0f) ? 1.0f : 1.0f / v;
        }

        // out[n, b, h] = oacc / rowsum   ([B,N,H] -> [N,B,H] transpose)
#pragma unroll
        for (int c = 0; c < 4; ++c)
#pragma unroll
            for (int j = 0; j < 8; ++j)
                out[((size_t)(n0 + half * 8 + j) * kB + b) * kH + c * 16 + r] =
                    oacc[c][j] * invv[j];

        // Rescale this tile's A rows in place with b128 ops. Store->load from
        // the same wave is kept in order (ISA 7.3) and the data is L2-hot.
#pragma unroll
        for (int j = 0; j < 8; ++j) {
            const size_t base = ((size_t)b * kN + n0 + half * 8 + j) * (size_t)kN;
            const float iv = invv[j];
            for (int m4 = r * 4; m4 < kN; m4 += 64) {
                float4* p = (float4*)(Aout + base + m4);
                float4 x = *p;
                x.x *= iv; x.y *= iv; x.z *= iv; x.w *= iv;
                *p = x;
            }
        }
    }
}

// ---------------------------------------------------------------------------
extern "C" void kernel_launch(void* const* d_in, const int* in_sizes, int n_in,
                              void* d_out, int out_size, void* d_ws, size_t ws_size,
                              hipStream_t stream) {
    (void)in_sizes; (void)n_in; (void)out_size; (void)ws_size;
    const float* q    = (const float*)d_in[0];
    const float* k    = (const float*)d_in[1];
    const float* v    = (const float*)d_in[2];
    const float* mask = (const float*)d_in[3];
    const float* Wq   = (const float*)d_in[4];
    const float* bq   = (const float*)d_in[5];
    const float* Wk   = (const float*)d_in[6];
    const float* bk   = (const float*)d_in[7];
    const float* Wv   = (const float*)d_in[8];
    const float* bv   = (const float*)d_in[9];

    const size_t mat_elems = (size_t)kB * kN * kH;  // 4,194,304
    float* qp = (float*)d_ws;
    float* kp = qp + mat_elems;
    float* vp = kp + mat_elems;

    float* out  = (float*)d_out;                 // [N,B,H]
    float* Aout = out + mat_elems;               // [B,N,N] (N*B*H == B*N*H)

    proj_kernel<<<512, 256, 0, stream>>>(q, Wq, bq, qp);
    proj_kernel<<<512, 256, 0, stream>>>(k, Wk, bk, kp);
    proj_kernel<<<512, 256, 0, stream>>>(v, Wv, bv, vp);
    attn_kernel<<<128, 256, 0, stream>>>(qp, kp, vp, mask, out, Aout);
}